// MyLSTM_87849261072496
// MI455X (gfx1250) — compile-verified
//
#include <hip/hip_runtime.h>
#include <hip/hip_bf16.h>

// ---------------------------------------------------------------------------
// LSTM (B=4096, T=512, F=64, H=3) + dense sigmoid head, fp32.
// Phase 1: xz[t][b][12] = x[b][t][:] @ W + bias  via V_WMMA_F32_16X16X4_F32
// Phase 2: per-lane recurrence over T (one lane per batch chain),
//          packed-f32 gate FMAs, exp/rcp-based activations.
// Chunked over T so the xz staging buffer fits ws_size and stays L2-resident.
// ---------------------------------------------------------------------------

#define LSTM_B 4096
#define LSTM_T 512
#define LSTM_F 64
#define LSTM_G 12   // 4*H

typedef __attribute__((ext_vector_type(2))) float v2f;
typedef __attribute__((ext_vector_type(4))) float v4f;
typedef __attribute__((ext_vector_type(8))) float v8f;

// ---------------- Phase 1: input projection with fp32 WMMA ------------------
// grid: tc*32 blocks of 256 threads = tc*256 waves; each wave: one t, 16 b-rows.
__global__ __launch_bounds__(256) void lstm_proj_wmma(
    const float* __restrict__ x,     // [B, T, F]
    const float* __restrict__ W,     // [F, 12]
    const float* __restrict__ bias,  // [12]
    float* __restrict__ xz,          // [tc, B, 12] (chunk-local)
    int t0)
{
    const int lane       = threadIdx.x & 31;
    const int waveGlobal = blockIdx.x * 8 + (threadIdx.x >> 5);
    const int tt         = waveGlobal >> 8;          // 0..tc-1  (256 waves per t)
    const int t          = t0 + tt;
    const int b0         = (waveGlobal & 255) << 4;  // 16 batch rows per wave
    const int m          = lane & 15;                // M index (row within tile)
    const int half       = lane >> 4;                // K-pair select (ISA A layout)
    const int col        = m;                        // N index for B/D fragments

    // B fragment: W column `col` (zero-padded for col >= 12), all K in registers.
    // f32 B 4x16 layout: VGPR0 = {K=0 | K=2}, VGPR1 = {K=1 | K=3} per lane N.
    v2f bfrag[16];
#pragma unroll
    for (int kk = 0; kk < 16; ++kk) {
        const int k = kk * 4 + half * 2;
        float w0 = 0.0f, w1 = 0.0f;
        if (col < LSTM_G) {
            w0 = W[k * LSTM_G + col];
            w1 = W[(k + 1) * LSTM_G + col];
        }
        bfrag[kk].x = w0;
        bfrag[kk].y = w1;
    }

    // A fragment source: row (b0+m) of x at time t; lane's K-pair offset = half*2.
    const float* xrow = x + ((size_t)(b0 + m) * LSTM_T + t) * LSTM_F + half * 2;
    const v2f* xrow2  = (const v2f*)xrow;            // 8B aligned (x rows 256B aligned)

    v8f acc = {0.f, 0.f, 0.f, 0.f, 0.f, 0.f, 0.f, 0.f};
#pragma unroll
    for (int kk = 0; kk < 16; ++kk) {
        // Non-temporal: 512MB stream of x must not evict xz from L2.
        v2f a = __builtin_nontemporal_load(xrow2 + 2 * kk);
        acc = __builtin_amdgcn_wmma_f32_16x16x4_f32(
            false, a, false, bfrag[kk], (short)0, acc, false, false);
    }

    // Store D (+bias). D layout: VGPR r -> row r (lanes 0-15) / row 8+r (lanes 16-31).
    if (col < LSTM_G) {
        const float bs = bias[col];
        float* outp = xz + ((size_t)tt * LSTM_B + b0) * LSTM_G + col;
#pragma unroll
        for (int r = 0; r < 8; ++r) {
            const int row = r + half * 8;
            outp[row * LSTM_G] = acc[r] + bs;   // regular store: keep L2-resident
        }
    }
}

// ---------------- Phase 2: recurrence, one lane per batch chain -------------
__device__ __forceinline__ float fast_sigmoid(float z) {
    float e = __expf(-z);
    return __builtin_amdgcn_rcpf(1.0f + e);
}
// tanh(x) = 2*sigmoid(2x) - 1; saturates cleanly (exp->inf => rcp->0 => -1).
__device__ __forceinline__ float fast_tanh(float z) {
    return fmaf(2.0f, fast_sigmoid(2.0f * z), -1.0f);
}

__global__ __launch_bounds__(256) void lstm_rec(
    const float* __restrict__ xz,    // [tc, B, 12]
    const float* __restrict__ U,     // [3, 12]
    const float* __restrict__ Wd,    // [3]
    const float* __restrict__ bd,    // [1]
    float* __restrict__ carry,       // [B, 6]  (h0..h2, c0..c2)
    float* __restrict__ out,         // [B]
    int t0, int tc)
{
    const int b = blockIdx.x * 256 + threadIdx.x;   // 0..B-1

    // U is wave-uniform -> scalar loads; pack as float2 pairs for v_pk_fma_f32.
    v2f Ur[3][6];
#pragma unroll
    for (int j = 0; j < 3; ++j)
#pragma unroll
        for (int p = 0; p < 6; ++p) {
            Ur[j][p].x = U[j * LSTM_G + 2 * p];
            Ur[j][p].y = U[j * LSTM_G + 2 * p + 1];
        }

    float h[3], c[3];
    if (t0 == 0) {
#pragma unroll
        for (int u = 0; u < 3; ++u) { h[u] = 0.0f; c[u] = 0.0f; }
    } else {
#pragma unroll
        for (int u = 0; u < 3; ++u) {
            h[u] = carry[b * 6 + u];
            c[u] = carry[b * 6 + 3 + u];
        }
    }

    const v4f* zp = (const v4f*)(xz + (size_t)b * LSTM_G);  // 16B aligned (48B rows)
    const size_t zstride4 = (size_t)LSTM_B * LSTM_G / 4;    // v4f elems per t-slice

    for (int tt = 0; tt < tc; ++tt) {
        // Read-once stream: NT hint frees L2 lines for the next proj chunk.
        v4f z0 = __builtin_nontemporal_load(zp + 0);
        v4f z1 = __builtin_nontemporal_load(zp + 1);
        v4f z2 = __builtin_nontemporal_load(zp + 2);
        zp += zstride4;

        // z (12 gates) as 6 packed pairs: z += h0*U0 + h1*U1 + h2*U2 (v_pk_fma_f32)
        v2f zv[6] = {{z0.x, z0.y}, {z0.z, z0.w}, {z1.x, z1.y},
                     {z1.z, z1.w}, {z2.x, z2.y}, {z2.z, z2.w}};
#pragma unroll
        for (int j = 0; j < 3; ++j) {
            const v2f hb = {h[j], h[j]};
#pragma unroll
            for (int p = 0; p < 6; ++p)
                zv[p] = __builtin_elementwise_fma(hb, Ur[j][p], zv[p]);
        }
        const float z[12] = {zv[0].x, zv[0].y, zv[1].x, zv[1].y,
                             zv[2].x, zv[2].y, zv[3].x, zv[3].y,
                             zv[4].x, zv[4].y, zv[5].x, zv[5].y};
#pragma unroll
        for (int u = 0; u < 3; ++u) {
            const float ig = fast_sigmoid(z[u]);
            const float fg = fast_sigmoid(z[3 + u]);
            const float gg = fast_tanh(z[6 + u]);
            const float og = fast_sigmoid(z[9 + u]);
            c[u] = fmaf(fg, c[u], ig * gg);
            h[u] = og * fast_tanh(c[u]);
        }
    }

    if (t0 + tc == LSTM_T) {
        float acc = bd[0];
#pragma unroll
        for (int u = 0; u < 3; ++u) acc = fmaf(h[u], Wd[u], acc);
        out[b] = fast_sigmoid(acc);
    } else {
#pragma unroll
        for (int u = 0; u < 3; ++u) {
            carry[b * 6 + u]     = h[u];
            carry[b * 6 + 3 + u] = c[u];
        }
    }
}

// ---------------------------------------------------------------------------
extern "C" void kernel_launch(void* const* d_in, const int* in_sizes, int n_in,
                              void* d_out, int out_size, void* d_ws, size_t ws_size,
                              hipStream_t stream) {
    const float* x    = (const float*)d_in[0];
    const float* W    = (const float*)d_in[1];
    const float* U    = (const float*)d_in[2];
    const float* bias = (const float*)d_in[3];
    const float* Wd   = (const float*)d_in[4];
    const float* bd   = (const float*)d_in[5];
    float* out        = (float*)d_out;

    // Workspace layout: [carry: B*6 floats (256B padded)] [xz chunk: tc*B*12 floats]
    const size_t carryBytes = (size_t)LSTM_B * 6 * sizeof(float);
    const size_t carryPad   = (carryBytes + 255) & ~(size_t)255;
    float* carry = (float*)d_ws;
    float* xzbuf = (float*)((char*)d_ws + carryPad);

    const size_t sliceBytes = (size_t)LSTM_B * LSTM_G * sizeof(float);  // per-t slice
    size_t avail = (ws_size > carryPad) ? (ws_size - carryPad) : 0;
    int cap = (int)(avail / sliceBytes);
    int Tc = cap < 1 ? 1 : cap;
    if (Tc > 128) Tc = 128;   // <=24MiB staging: stays resident in 192MB L2

    for (int t0 = 0; t0 < LSTM_T; t0 += Tc) {
        const int tc = (LSTM_T - t0) < Tc ? (LSTM_T - t0) : Tc;
        // tc*256 waves -> tc*32 blocks of 256 threads
        lstm_proj_wmma<<<tc * 32, 256, 0, stream>>>(x, W, bias, xzbuf, t0);
        lstm_rec<<<LSTM_B / 256, 256, 0, stream>>>(xzbuf, U, Wd, bd, carry, out, t0, tc);
    }
}